// Pooling_14336600834463
// MI455X (gfx1250) — compile-verified
//
#include <hip/hip_runtime.h>
#include <stdint.h>

typedef float v2f __attribute__((ext_vector_type(2)));
typedef float v4f __attribute__((ext_vector_type(4)));
typedef float v8f __attribute__((ext_vector_type(8)));

#define Bn    4
#define Tn    1024
#define Dn    512
#define TBLK  8
#define TROWS 42           /* staged absolute-time rows (e_max=6 + 36 K slots) */
#define TSTR  44           /* LDS time-stride per d column (42 + pad, bank-safe) */
#define DCHK  256          /* d-columns staged per block (half of D) */
#define EPSV  1e-12f

// ---------------------------------------------------------------------------
// Pass 1: per-row reciprocal L2 norms. One wave per (b,t) row of 512 floats.
// ---------------------------------------------------------------------------
__global__ __launch_bounds__(256) void norm_kernel(const float* __restrict__ x,
                                                   float* __restrict__ rnorm) {
    const int wave = threadIdx.x >> 5;
    const int lane = threadIdx.x & 31;
    const int row  = blockIdx.x * 8 + wave;            // 0..4095 = b*T + t
    const float* p = x + (size_t)row * Dn;
    float s = 0.0f;
#pragma unroll
    for (int j = 0; j < 4; ++j) {
        v4f v = *(const v4f*)(p + (size_t)(lane + 32 * j) * 4);
        s += v.x * v.x + v.y * v.y + v.z * v.z + v.w * v.w;
    }
#pragma unroll
    for (int m = 16; m >= 1; m >>= 1) s += __shfl_xor(s, m, 32);
    if (lane == 0) rnorm[row] = 1.0f / fmaxf(sqrtf(s), EPSV);
}

// ---------------------------------------------------------------------------
// Pass 2: banded-Toeplitz WMMA, transposed [d][time] LDS.
// Block = (dhalf, t-block of 8, b). Wave w owns t = t0 + w, K-origin aligned
// to e = w & ~1 so every B pair (K, K+1) is one aligned ds_load_b64.
// K = 36 slots (9 k-steps, unconditional); banded A (rnorm folded in) zeroes
// all pad slots, so the 9th WMMA contributes 0 for even waves.
// Staging: per-lane transposing async global->LDS b32 copies.
// ---------------------------------------------------------------------------
__global__ __launch_bounds__(256) void pool_wmma_kernel(
    const float* __restrict__ x,
    const float* __restrict__ W0, const float* __restrict__ W1,
    const float* __restrict__ W2, const float* __restrict__ W3,
    const float* __restrict__ rnorm, float* __restrict__ out) {

    __shared__ float lxt[DCHK * TSTR];   // [d][time], 45 KB
    __shared__ float lw[240];
    __shared__ float lrn[TROWS];

    const int dhalf = blockIdx.x;   // 0..1
    const int tb    = blockIdx.y;   // 0..127
    const int bb    = blockIdx.z;   // 0..3
    const int tid   = threadIdx.x;
    const int lane  = tid & 31;
    const int wave  = tid >> 5;

    const int t0    = tb * TBLK;
    const int dbase = dhalf * DCHK;

    // Stage W (concatenated row-major: offsets 0,16,48,112; 240 floats total).
    if (tid < 16)       lw[tid] = W0[tid];
    else if (tid < 48)  lw[tid] = W1[tid - 16];
    else if (tid < 112) lw[tid] = W2[tid - 48];
    else if (tid < 240) lw[tid] = W3[tid - 112];

    // Stage the rnorm window (one value per absolute-time row).
    if (tid < TROWS) {
        const int j = t0 - 15 + tid;
        lrn[tid] = (j >= 0 && j < Tn) ? rnorm[bb * Tn + j] : 0.0f;
    }

    // Transposing stage: thread 'tid' owns d-column (dbase + tid).
    // Row r holds absolute time j = t0-15+r; valid rows via per-lane async b32
    // (global reads coalesce across lanes), OOB rows zeroed through DS.
    {
        uint32_t lo = (uint32_t)(uintptr_t)&lxt[tid * TSTR];          // += 4/row
        uint32_t go = (uint32_t)(((uint32_t)(bb * Tn + (t0 - 15)) * Dn +
                                  dbase + tid) * sizeof(float));      // += 2048/row
        for (int r = 0; r < TROWS; ++r) {
            const int j = t0 - 15 + r;            // block-uniform validity
            if (j >= 0 && j < Tn) {
                asm volatile("global_load_async_to_lds_b32 %0, %1, %2"
                             :: "v"(lo), "v"(go), "s"(x) : "memory");
            } else {
                lxt[tid * TSTR + r] = 0.0f;
            }
            lo += (uint32_t)sizeof(float);
            go += (uint32_t)(Dn * sizeof(float));
        }
    }
    asm volatile("s_wait_asynccnt 0x0" ::: "memory");
    __syncthreads();

    // This wave's output timestep and even-aligned K origin.
    const int t    = t0 + wave;
    const int e    = wave & ~1;          // K slot 0 -> LDS row e
    const int hi   = lane >> 4;          // half-wave (selects K pair)
    const int mrow = lane & 15;          // A-matrix M row = scale*4 + head
    const int sc   = mrow >> 2;
    const int oo   = mrow & 3;
    const int s    = 4 << sc;
    const int start = max(0, t - (s >> 1) + 1);
    const int last  = min(t + (s >> 1), Tn - 1);

    // Branchless banded A with rnorm[j] folded in: lane holds A[mrow][K],
    // K = 4k + 2*hi + v  ->  LDS row rr = e + K,  j = t0-15+rr.
    // woff = 16*((1<<sc)-1) in {0,16,48,112};  oo*s = oo << (2+sc).
    const int wbase = 16 * ((1 << sc) - 1) + (oo << (2 + sc));
    v2f A[9];
#pragma unroll
    for (int k = 0; k < 9; ++k) {
#pragma unroll
        for (int v = 0; v < 2; ++v) {
            const int rr = e + k * 4 + 2 * hi + v;
            const int j  = t0 - 15 + rr;
            const int ki = min(max(j - start, 0), s - 1);   // clamped, in-bounds
            const float wv = lw[wbase + ki] * lrn[rr];      // unconditional loads
            A[k][v] = (j >= start && j <= last) ? wv : 0.0f; // -> v_cndmask
        }
    }

    const int n = lane & 15;             // N (d) index for B and D tiles
    // Element offsets of the two scale-planes this half-wave writes:
    // m = r + 8*hi -> msc = 2*hi + (r>>2), mo = r&3; store at vo + mo*Dn.
    uint32_t vo0 = (uint32_t)(((2 * hi)     * (Tn * Bn) + t * Bn + bb) * 4) * Dn
                 + (uint32_t)(dbase + n);
    uint32_t vo1 = (uint32_t)(((2 * hi + 1) * (Tn * Bn) + t * Bn + bb) * 4) * Dn
                 + (uint32_t)(dbase + n);

    for (int dc = 0; dc < DCHK / 16; ++dc) {
        // Pair base: rows (e + 4k + 2hi, +1) are adjacent floats -> ds_load_b64
        const float* bp = &lxt[(dc * 16 + n) * TSTR + e + 2 * hi];
        v8f c = {0.0f, 0.0f, 0.0f, 0.0f, 0.0f, 0.0f, 0.0f, 0.0f};
#pragma unroll
        for (int k = 0; k < 9; ++k) {
            v2f Bv = *(const v2f*)(bp + 4 * k);
            c = __builtin_amdgcn_wmma_f32_16x16x4_f32(
                    false, A[k], false, Bv, (short)0, c, false, false);
        }
        // Streaming (non-temporal) stores; immediates fold mo*Dn*4 = 0..6144B.
#pragma unroll
        for (int r = 0; r < 4; ++r)
            __builtin_nontemporal_store(c[r],     out + vo0 + r * Dn);
#pragma unroll
        for (int r = 0; r < 4; ++r)
            __builtin_nontemporal_store(c[r + 4], out + vo1 + r * Dn);
        vo0 += 16; vo1 += 16;
    }
}

// ---------------------------------------------------------------------------
extern "C" void kernel_launch(void* const* d_in, const int* in_sizes, int n_in,
                              void* d_out, int out_size, void* d_ws, size_t ws_size,
                              hipStream_t stream) {
    (void)in_sizes; (void)n_in; (void)out_size; (void)ws_size;
    const float* x  = (const float*)d_in[0];
    const float* W0 = (const float*)d_in[1];
    const float* W1 = (const float*)d_in[2];
    const float* W2 = (const float*)d_in[3];
    const float* W3 = (const float*)d_in[4];
    float* out   = (float*)d_out;
    float* rnorm = (float*)d_ws;         // 4096 floats of scratch

    norm_kernel<<<dim3(512), dim3(256), 0, stream>>>(x, rnorm);
    pool_wmma_kernel<<<dim3(2, 128, 4), dim3(256), 0, stream>>>(
        x, W0, W1, W2, W3, rnorm, out);
}